// MPS_pytorch_sharing_83880711290966
// MI455X (gfx1250) — compile-verified
//
#include <hip/hip_runtime.h>

// MPS chain contraction for MI455X (gfx1250).
// 32768 chains x 221 sequential 3x3 steps. ~87 MB streamed -> ~3.8 us at 23.3 TB/s.
// Per-wave TDM (tensor_load_to_lds) double-buffered staging: each wave32 DMAs
// its own 32 chains' tile (32 rows x 48 dwords, stride 693 dwords) into LDS and
// synchronizes only via its own TENSORcnt -- no workgroup barriers, so the
// shared T coefficients stay on the scalar (s_load) path.

#define NSTEP    221
#define ROWF     693          // floats per chain row (231 positions * 3)
#define TPB      256
#define NCHAIN   32768
#define TILE     16           // steps per tile
#define NTILES   14           // 13 full tiles + 1 partial (13 steps)
#define TILE_DW  48           // dwords per chain per tile (TILE*3)
#define LROW_DW  51           // LDS row stride: 48 data + 3 pad dwords (1 per 16)
#define WROWS    32           // chains (rows) per wave
#define LDS_DW   (LROW_DW * TPB)   // 13056 dwords per buffer

typedef unsigned int       u32;
typedef unsigned long long u64;
typedef u32 u32x4 __attribute__((ext_vector_type(4)));
typedef int i32x4 __attribute__((ext_vector_type(4)));
typedef int i32x8 __attribute__((ext_vector_type(8)));

#if defined(__has_include)
#if __has_include(<hip/amd_detail/amd_gfx1250_TDM.h>)
#define TDM_ARITY6 1
#else
#define TDM_ARITY6 0
#endif
#else
#define TDM_ARITY6 0
#endif

__device__ __forceinline__ void mps_step(float& v0, float& v1, float& v2,
                                         float x0, float x1, float x2,
                                         const float* __restrict__ tl) {
  // p[l][d] = v[l] * x[d]
  float p00 = v0 * x0, p01 = v0 * x1, p02 = v0 * x2;
  float p10 = v1 * x0, p11 = v1 * x1, p12 = v1 * x2;
  float p20 = v2 * x0, p21 = v2 * x1, p22 = v2 * x2;
  // vec_new[r] = vec[r] + sum_{l,d} p[l][d] * T[l*9 + r*3 + d]
  float n0 = v0, n1 = v1, n2 = v2;
  n0 = fmaf(p00, tl[0],  n0);  n1 = fmaf(p00, tl[3],  n1);  n2 = fmaf(p00, tl[6],  n2);
  n0 = fmaf(p01, tl[1],  n0);  n1 = fmaf(p01, tl[4],  n1);  n2 = fmaf(p01, tl[7],  n2);
  n0 = fmaf(p02, tl[2],  n0);  n1 = fmaf(p02, tl[5],  n1);  n2 = fmaf(p02, tl[8],  n2);
  n0 = fmaf(p10, tl[9],  n0);  n1 = fmaf(p10, tl[12], n1);  n2 = fmaf(p10, tl[15], n2);
  n0 = fmaf(p11, tl[10], n0);  n1 = fmaf(p11, tl[13], n1);  n2 = fmaf(p11, tl[16], n2);
  n0 = fmaf(p12, tl[11], n0);  n1 = fmaf(p12, tl[14], n1);  n2 = fmaf(p12, tl[17], n2);
  n0 = fmaf(p20, tl[18], n0);  n1 = fmaf(p20, tl[21], n1);  n2 = fmaf(p20, tl[24], n2);
  n0 = fmaf(p21, tl[19], n0);  n1 = fmaf(p21, tl[22], n1);  n2 = fmaf(p21, tl[25], n2);
  n0 = fmaf(p22, tl[20], n0);  n1 = fmaf(p22, tl[23], n1);  n2 = fmaf(p22, tl[26], n2);
  v0 = n0; v1 = n1; v2 = n2;
}

#if defined(__AMDGCN__) && __has_builtin(__builtin_amdgcn_tensor_load_to_lds)
#define USE_TDM 1
#else
#define USE_TDM 0
#endif

#if USE_TDM
// One per-wave TDM 2D tile: 32 rows (this wave's chains) x 48 dwords, global
// row stride 693 dwords, into LDS with 1 pad dword per 16 data dwords
// (51-dword LDS rows, gcd(51,64)=1 -> bank-conflict-free ds reads).
// Issued by every wave (one DMA per wave); descriptor args are wave-uniform.
__device__ __forceinline__ void tdm_issue(const float* samples, u32 chain0, u32 l0,
                                          u32 lds_byte_addr) {
  const u64 ga = (u64)(size_t)samples +
                 ((u64)chain0 * (u64)ROWF + (u64)l0 * 3u) * 4u;
  // D# group 0: count=1 | lds_addr | global_addr | type=2 ("image")
  u32x4 g0 = { 1u,
               lds_byte_addr,
               (u32)ga,
               (u32)(ga >> 32) | 0x80000000u };
  // D# group 1:
  //  d0: data_size=4B (2<<16) | pad_enable (1<<20) | pad_interval=16dw (3<<22) | pad_amount=1dw (0)
  //  tensor_dim0=693 (79:48), tensor_dim1=32 (111:80),
  //  tile_dim0=48 (127:112), tile_dim1=32 (143:128), tile_dim2=0,
  //  tensor_dim0_stride=693 (207:160), tensor_dim1_stride=0
  i32x8 g1 = { (int)((2u << 16) | (1u << 20) | (3u << 22)),
               (int)(((u32)ROWF & 0xFFFFu) << 16),
               (int)((((u32)ROWF >> 16) & 0xFFFFu) | ((u32)WROWS << 16)),
               (int)(((u32)WROWS >> 16) | ((u32)TILE_DW << 16)),
               (int)WROWS,
               (int)ROWF,
               0, 0 };
  i32x4 z4 = { 0, 0, 0, 0 };
#if TDM_ARITY6
  i32x8 z8 = { 0, 0, 0, 0, 0, 0, 0, 0 };
  __builtin_amdgcn_tensor_load_to_lds(g0, g1, z4, z4, z8, 0);
#else
  __builtin_amdgcn_tensor_load_to_lds(g0, g1, z4, z4, 0);
#endif
}

// Compute NS steps of one tile from this lane's padded LDS row.
template <int NS>
__device__ __forceinline__ void tile_compute(const float* __restrict__ T, int l0,
                                             const float* __restrict__ row,
                                             float& v0, float& v1, float& v2) {
#pragma unroll
  for (int u = 0; u < NS; ++u) {
    const int d0 = 3 * u,     i0 = d0 + (d0 >> 4);   // pad dword after every 16
    const int d1 = 3 * u + 1, i1 = d1 + (d1 >> 4);
    const int d2 = 3 * u + 2, i2 = d2 + (d2 >> 4);
    mps_step(v0, v1, v2, row[i0], row[i1], row[i2], T + (l0 + u) * 27);
  }
}
#endif  // USE_TDM

__global__ __launch_bounds__(TPB) void mps_chain_kernel(
    const float* __restrict__ samples,   // [32768][693]
    const float* __restrict__ T,         // [221][27], wave-uniform scalar reads
    float* __restrict__ out)             // [32768][3]
{
  const int chain = blockIdx.x * TPB + threadIdx.x;
  float v0 = 1.0f, v1 = 0.0f, v2 = 0.0f;

#if USE_TDM
  __shared__ float smem[2][LDS_DW];      // 2 x 51 KB double buffer (per-wave slices)
  // Wave-uniform wave id: TDM descriptors must be scalar, and the DMA is
  // issued once per wave (EXEC-independent), each wave staging its own rows.
  const u32 wave_id = (u32)__builtin_amdgcn_readfirstlane((int)(threadIdx.x >> 5));
  const u32 chain0  = (u32)blockIdx.x * TPB + wave_id * WROWS;  // this wave's chains
  const u32 lrow    = (u32)LROW_DW * threadIdx.x;               // this lane's LDS row
  const u32 wbase   = (u32)(WROWS * LROW_DW) * wave_id;         // wave slice base (dw)

  tdm_issue(samples, chain0, 0, (u32)(size_t)&smem[0][wbase]);

#pragma unroll 1
  for (int t = 0; t < NTILES - 1; ++t) {           // 13 full tiles
    // DMA tile t+1 into the other buffer (its last reader was tile t-1,
    // already finished in this wave's program order), then wait for tile t.
    tdm_issue(samples, chain0, (u32)(TILE * (t + 1)),
              (u32)(size_t)&smem[(t + 1) & 1][wbase]);
#if __has_builtin(__builtin_amdgcn_s_wait_tensorcnt)
    __builtin_amdgcn_s_wait_tensorcnt(1);          // in-order per wave: tile t landed
#endif
    const float* row = &smem[t & 1][lrow];
    tile_compute<TILE>(T, TILE * t, row, v0, v1, v2);
  }
  // tail tile: steps 208..220 (13 steps; TDM loaded 16, extras are in-bounds)
#if __has_builtin(__builtin_amdgcn_s_wait_tensorcnt)
  __builtin_amdgcn_s_wait_tensorcnt(0);
#endif
  {
    const float* row = &smem[(NTILES - 1) & 1][lrow];
    tile_compute<NSTEP - TILE * (NTILES - 1)>(T, TILE * (NTILES - 1), row, v0, v1, v2);
  }
#else
  // Fallback (also what the host pass sees): direct per-lane streaming with
  // near-cache prefetch; compiler merges the 8-step tiles into global_load_b128.
  const float* __restrict__ x = samples + (size_t)chain * ROWF;
#pragma unroll 1
  for (int tile = 0; tile < 27; ++tile) {
    const int lbase = tile * 8;
    int pl = lbase + 72;
    if (pl > NSTEP - 1) pl = NSTEP - 1;
    __builtin_prefetch(x + pl * 3, 0, 3);
#pragma unroll
    for (int u = 0; u < 8; ++u) {
      const int l = lbase + u;
      const float* xp = x + l * 3;
      mps_step(v0, v1, v2, xp[0], xp[1], xp[2], T + l * 27);
    }
  }
#pragma unroll
  for (int l = 216; l < NSTEP; ++l) {
    const float* xp = x + l * 3;
    mps_step(v0, v1, v2, xp[0], xp[1], xp[2], T + l * 27);
  }
#endif

  float* o = out + (size_t)chain * 3;
  o[0] = v0; o[1] = v1; o[2] = v2;
}

extern "C" void kernel_launch(void* const* d_in, const int* in_sizes, int n_in,
                              void* d_out, int out_size, void* d_ws, size_t ws_size,
                              hipStream_t stream) {
  (void)in_sizes; (void)n_in; (void)d_ws; (void)ws_size; (void)out_size;
  const float* samples = (const float*)d_in[0];   // [256,128,11,21,3] f32
  const float* T       = (const float*)d_in[1];   // [221,3,3,3] f32
  // d_in[2] is bias_mat == eye(3); identity folded into the step update.
  float* out = (float*)d_out;                     // [256,128,3] f32

  mps_chain_kernel<<<NCHAIN / TPB, TPB, 0, stream>>>(samples, T, out);
}